// ButterflyRotation_55456617726325
// MI455X (gfx1250) — compile-verified
//
#include <hip/hip_runtime.h>
#include <hip/hip_bf16.h>
#include <stdint.h>

// Butterfly O(d log d) rotation, d=4096, 12 layers, batch 8192.
// Memory-bound (AI ~4.5 flop/byte, HBM floor ~11us): optimize the data path.
// CDNA5 paths: TDM tensor_load_to_lds (TENSORcnt) double-buffers 16KB rows
// into LDS; 3x4-layer register-resident butterfly stages; coalesced reg->global
// stores. WMMA deliberately unused: dense 16x16 block formulation would 4x the
// FLOPs of the butterfly at f32 WMMA's weakest shape for zero bandwidth gain.

#define DIMX   4096
#define NPAIR  2048
#define NLAYER 12
#define TPB    256
#define ROWB   (DIMX * 4)          // bytes per row

typedef uint32_t v4u __attribute__((ext_vector_type(4)));
typedef uint32_t v8u __attribute__((ext_vector_type(8)));

// offset (within a 16-element butterfly group) of the p-th pair's LEFT element
// for local stride 2^l
__device__ __forceinline__ int left_of(int p, int l) {
  int s = 1 << l;
  return ((p >> l) << (l + 1)) | (p & (s - 1));
}
// angle index for layer l given the LEFT element index e (bit l of e deleted)
__device__ __forceinline__ int q_of(int e, int l) {
  int s = 1 << l;
  return ((e >> (l + 1)) << l) | (e & (s - 1));
}
// element index of the p-th LEFT element owned by thread t at layer l,
// under the 3-stage register layout.
__device__ __forceinline__ int left_elem(int l, int p, int t) {
  if (l < 4)  return (t << 4) + left_of(p, l);                              // A: e = 16t + j
  if (l < 8)  return ((t >> 4) << 8) + (left_of(p, l - 4) << 4) + (t & 15); // B: e = hi*256 + 16m + lo
  return (left_of(p, l - 8) << 8) + t;                                      // C: e = 256m + t
}

__device__ __forceinline__ void rotate_pair(float& xl, float& xr, float c, float s) {
  float l0 = xl, r0 = xr;
  xl = __builtin_fmaf(c, l0,  s * r0);   // c*xl + s*xr
  xr = __builtin_fmaf(c, r0, -s * l0);   // -s*xl + c*xr
}

// One TDM descriptor: DMA a full 16KB row (4096 x f32) from global into LDS.
// D# group0: count=1 | lds_addr | global_addr[56:0] | type=2
// D# group1: data_size=4B, tensor 4096x1, tile 4096x1, stride 4096, no cluster.
__device__ __forceinline__ void tdm_load_row(const float* x, int row, uint32_t lds_off) {
  uint64_t ga = (uint64_t)(uintptr_t)x + (uint64_t)((uint32_t)row * (uint32_t)ROWB);
  v4u g0;
  g0.x = 0x1u;                                   // count=1, is_restore=0, gather=0
  g0.y = lds_off;                                // lds_addr (bytes)
  g0.z = (uint32_t)ga;                           // global_addr[31:0]
  g0.w = ((uint32_t)(ga >> 32) & 0x01FFFFFFu)    // global_addr[56:32]
         | 0x80000000u;                          // type = 2 ("image")
  v8u g1;
  g1[0] = 0x00020000u;                           // wg_mask=0, data_size=2 (4B)
  g1[1] = (uint32_t)DIMX << 16;                  // tensor_dim0[15:0] = 4096
  g1[2] = 1u << 16;                              // tensor_dim0 hi=0, tensor_dim1[15:0]=1
  g1[3] = (uint32_t)DIMX << 16;                  // tensor_dim1 hi=0, tile_dim0=4096
  g1[4] = 1u;                                    // tile_dim1=1, tile_dim2=0
  g1[5] = (uint32_t)DIMX;                        // tensor_dim0_stride[31:0]=4096
  g1[6] = (uint32_t)DIMX << 16;                  // stride hi=0, tensor_dim1_stride lo=4096
  g1[7] = 0u;
  asm volatile("tensor_load_to_lds %0, %1, null, null"
               :: "s"(g0), "s"(g1) : "memory");
}

__global__ __launch_bounds__(TPB) void butterfly_rot_kernel(
    const float* __restrict__ x, const float* __restrict__ angles,
    float* __restrict__ out, int batch)
{
  __shared__ float buf[2][DIMX];         // 32 KB double buffer
  const int t = threadIdx.x;
  const bool wave0 = (t < 32);           // wave32: lane-exact first wave

  // ---- per-thread rotation coefficients (row-invariant -> VGPRs) ----
  float2 csv[NLAYER][8];
  #pragma unroll
  for (int l = 0; l < NLAYER; ++l) {
    #pragma unroll
    for (int p = 0; p < 8; ++p) {
      int e = left_elem(l, p, t);
      int q = q_of(e, l);
      float a = angles[l * NPAIR + q];
      float sn, cn;
      __sincosf(a, &sn, &cn);
      csv[l][p].x = cn;
      csv[l][p].y = sn;
    }
  }

  // LDS byte offsets of the two buffers (flat addr low 32 bits == LDS offset)
  const uint32_t lds0 = (uint32_t)(uintptr_t)(&buf[0][0]);
  const uint32_t lds1 = (uint32_t)(uintptr_t)(&buf[1][0]);

  int row = (int)blockIdx.x;
  if (row < batch && wave0) tdm_load_row(x, row, lds0);

  int cur = 0;
  for (; row < batch; row += (int)gridDim.x, cur ^= 1) {
    // wave0's outstanding TDM (current row) completes, then make visible to all
    if (wave0) __builtin_amdgcn_s_wait_tensorcnt(0);
    __syncthreads();

    // overlap: DMA next row into the other buffer while we compute
    int nrow = row + (int)gridDim.x;
    if (nrow < batch && wave0) tdm_load_row(x, nrow, cur ? lds0 : lds1);

    float* b = buf[cur];
    float r16[16];

    // ---------- stage A: 16 contiguous elements, layers 0..3 ----------
    float4* my4 = (float4*)(b + (t << 4));
    #pragma unroll
    for (int j = 0; j < 4; ++j) ((float4*)r16)[j] = my4[j];
    #pragma unroll
    for (int l = 0; l < 4; ++l) {
      int s = 1 << l;
      #pragma unroll
      for (int p = 0; p < 8; ++p) {
        int jl = left_of(p, l);
        rotate_pair(r16[jl], r16[jl + s], csv[l][p].x, csv[l][p].y);
      }
    }
    // write back own region (exclusive ownership: no barrier needed before write)
    #pragma unroll
    for (int j = 0; j < 4; ++j) my4[j] = ((float4*)r16)[j];
    __syncthreads();

    // ---------- stage B: elements hi*256 + 16m + lo, layers 4..7 ----------
    const int hi = t >> 4, lo = t & 15;
    #pragma unroll
    for (int m = 0; m < 16; ++m) r16[m] = b[(hi << 8) + (m << 4) + lo];
    #pragma unroll
    for (int l = 4; l < 8; ++l) {
      int s = 1 << (l - 4);
      #pragma unroll
      for (int p = 0; p < 8; ++p) {
        int ml = left_of(p, l - 4);
        rotate_pair(r16[ml], r16[ml + s], csv[l][p].x, csv[l][p].y);
      }
    }
    #pragma unroll
    for (int m = 0; m < 16; ++m) b[(hi << 8) + (m << 4) + lo] = r16[m];
    __syncthreads();

    // ---------- stage C: elements 256m + t, layers 8..11 ----------
    #pragma unroll
    for (int m = 0; m < 16; ++m) r16[m] = b[(m << 8) + t];
    #pragma unroll
    for (int l = 8; l < 12; ++l) {
      int s = 1 << (l - 8);
      #pragma unroll
      for (int p = 0; p < 8; ++p) {
        int ml = left_of(p, l - 8);
        rotate_pair(r16[ml], r16[ml + s], csv[l][p].x, csv[l][p].y);
      }
    }

    // store straight from registers, coalesced b32 (STOREcnt, drained at endpgm)
    float* orow = out + (size_t)row * DIMX;
    #pragma unroll
    for (int m = 0; m < 16; ++m) orow[(m << 8) + t] = r16[m];
  }
}

extern "C" void kernel_launch(void* const* d_in, const int* in_sizes, int n_in,
                              void* d_out, int out_size, void* d_ws, size_t ws_size,
                              hipStream_t stream) {
  (void)n_in; (void)d_ws; (void)ws_size; (void)out_size;
  const float* x      = (const float*)d_in[0];
  const float* angles = (const float*)d_in[1];
  float* out          = (float*)d_out;

  int batch = in_sizes[0] / DIMX;
  int grid  = batch < 1024 ? batch : 1024;
  if (grid < 1) grid = 1;
  butterfly_rot_kernel<<<grid, TPB, 0, stream>>>(x, angles, out, batch);
}